// ChamferDistance_29094108463560
// MI455X (gfx1250) — compile-verified
//
#include <hip/hip_runtime.h>
#include <hip/hip_bf16.h>
#include <math.h>

typedef __attribute__((ext_vector_type(2))) float v2f;
typedef __attribute__((ext_vector_type(8))) float v8f;

#define CH_N   4
#define CH_P1  8192
#define CH_P2  8192
#define ROWS_PER_BLOCK 128   // 8 waves * 16 rows
#define THREADS 256
#define FLT_BIG 3.0e38f
#define ENC_INF 0xFF800000u  // order-flip encoding of +inf

// Order-preserving float -> uint map (total order over all finite floats):
//   positives -> [0x80000000, ...) increasing; negatives -> [0, 0x80000000) increasing
__device__ __forceinline__ unsigned flt_flip(float f) {
    unsigned u = __float_as_uint(f);
    return u ^ (unsigned)(((int)u >> 31) | 0x80000000);
}
__device__ __forceinline__ float flt_unflip(unsigned e) {
    unsigned u = (e & 0x80000000u) ? (e ^ 0x80000000u) : ~e;
    return __uint_as_float(u);
}

// ---------------------------------------------------------------------------
// Pre-pass: pack B tiles + column norms, init colmin.
//   packLo[p] = {-2*y0, -2*y1, ysq}   (consumed by lanes 0-15:  K=0,1)
//   packHi[p] = {-2*y2,  1.0,  ysq}   (consumed by lanes 16-31: K=2,3)
// ---------------------------------------------------------------------------
__global__ void chamfer_pack_y(const float* __restrict__ y,
                               float* __restrict__ packLo,
                               float* __restrict__ packHi,
                               unsigned* __restrict__ colmin) {
    int i = blockIdx.x * blockDim.x + threadIdx.x;   // over N*P2
    if (i < CH_N * CH_P2) {
        const float y0 = y[i * 3 + 0];
        const float y1 = y[i * 3 + 1];
        const float y2 = y[i * 3 + 2];
        const float ysq = y0 * y0 + y1 * y1 + y2 * y2;
        packLo[i * 3 + 0] = -2.0f * y0;
        packLo[i * 3 + 1] = -2.0f * y1;
        packLo[i * 3 + 2] = ysq;
        packHi[i * 3 + 0] = -2.0f * y2;
        packHi[i * 3 + 1] = 1.0f;
        packHi[i * 3 + 2] = ysq;
        colmin[i] = ENC_INF;
    }
}

// ---------------------------------------------------------------------------
// Main tile kernel. grid = (P1/128, N), block = 256 threads (8 waves).
// Wave w owns rows [blockIdx.x*128 + 16w, +16) of batch blockIdx.y and sweeps
// all P2 columns in 16-wide tiles using V_WMMA_F32_16X16X4_F32.
//
// A (16x4, lane = row M = lane&15): lanes 0-15 {x0,x1}; lanes 16-31 {x2,||x||^2}
// B (4x16,  lane = col N = lane&15): from packLo / packHi (half-selected base)
// => C[m][n] = ||x_m||^2 - 2 x_m.y_n ; add ysq (lane scalar) after.
// C/D layout: lane = N; VGPR r holds M=r (lanes 0-15), M=r+8 (lanes 16-31)
// ---------------------------------------------------------------------------
__global__ void chamfer_tiles(const float* __restrict__ x,
                              const float* __restrict__ packLo,
                              const float* __restrict__ packHi,
                              float*       __restrict__ rowmin,
                              unsigned*    __restrict__ colmin) {
    __shared__ unsigned ldsCol[CH_P2];   // 32 KB of the 320 KB/WGP LDS

    const int tid  = threadIdx.x;
    const int n    = blockIdx.y;
    const int lane = tid & 31;
    const int wave = tid >> 5;

    for (int i = tid; i < CH_P2; i += THREADS) ldsCol[i] = ENC_INF;
    __syncthreads();

    const float* xb = x + (size_t)n * CH_P1 * 3;
    const float* pb = ((lane < 16) ? packLo : packHi) + (size_t)n * CH_P2 * 3;

    const int rowBase = blockIdx.x * ROWS_PER_BLOCK + wave * 16;
    const int myRow   = rowBase + (lane & 15);

    // Build A tile once (lives in registers for the whole sweep)
    const float ax0 = xb[myRow * 3 + 0];
    const float ax1 = xb[myRow * 3 + 1];
    const float ax2 = xb[myRow * 3 + 2];
    const float axs = ax0 * ax0 + ax1 * ax1 + ax2 * ax2;
    v2f a;
    a.x = (lane < 16) ? ax0 : ax2;
    a.y = (lane < 16) ? ax1 : axs;

    v8f accMin;
#pragma unroll
    for (int r = 0; r < 8; ++r) accMin[r] = FLT_BIG;

    for (int c0 = 0; c0 < CH_P2; c0 += 16) {
        const int col = c0 + (lane & 15);
        // One b96 load per lane: {b.x, b.y, ysq} — zero VALU to build B
        const float b0  = pb[col * 3 + 0];
        const float b1  = pb[col * 3 + 1];
        const float ysq = pb[col * 3 + 2];
        v2f b; b.x = b0; b.y = b1;

        v8f c = {};
        c = __builtin_amdgcn_wmma_f32_16x16x4_f32(
                /*neg_a=*/false, a, /*neg_b=*/false, b,
                /*c_mod=*/(short)0, c, /*reuse_a=*/false, /*reuse_b=*/false);

        float cmin = FLT_BIG;
#pragma unroll
        for (int r = 0; r < 8; ++r) {
            const float v = c[r] + ysq;      // squared distance (may be ~-eps)
            accMin[r] = fminf(accMin[r], v); // row-direction running min
            cmin = fminf(cmin, v);           // col-direction partial (8 Ms)
        }
        // order-preserving encode -> native ds_min_u32 handles any sign
        atomicMin(&ldsCol[col], flt_flip(cmin));
    }

    // Row minima: reduce across the 16 lanes of each half (M fixed per r+half)
#pragma unroll
    for (int r = 0; r < 8; ++r) {
        float v = accMin[r];
#pragma unroll
        for (int s = 1; s < 16; s <<= 1)
            v = fminf(v, __shfl_xor(v, s, 32));
        accMin[r] = v;
    }
    if ((lane & 15) == 0) {
        const int mOff = (lane < 16) ? 0 : 8;
#pragma unroll
        for (int r = 0; r < 8; ++r)
            rowmin[(size_t)n * CH_P1 + rowBase + mOff + r] = accMin[r];
    }

    // Flush per-block column minima to global
    __syncthreads();
    for (int i = tid; i < CH_P2; i += THREADS)
        atomicMin(&colmin[(size_t)n * CH_P2 + i], ldsCol[i]);
}

// ---------------------------------------------------------------------------
// Finalize: out[n] = mean_p sqrt(max(rowmin,0)) + mean_q sqrt(max(colmin,0))
// ---------------------------------------------------------------------------
__global__ void chamfer_finalize(const float* __restrict__ rowmin,
                                 const unsigned* __restrict__ colmin,
                                 float* __restrict__ out) {
    const int n   = blockIdx.x;
    const int tid = threadIdx.x;
    float sx = 0.0f, sy = 0.0f;
    for (int i = tid; i < CH_P1; i += blockDim.x)
        sx += sqrtf(fmaxf(rowmin[(size_t)n * CH_P1 + i], 0.0f));
    for (int i = tid; i < CH_P2; i += blockDim.x)
        sy += sqrtf(fmaxf(flt_unflip(colmin[(size_t)n * CH_P2 + i]), 0.0f));

    __shared__ float redx[256];
    __shared__ float redy[256];
    redx[tid] = sx; redy[tid] = sy;
    __syncthreads();
    for (int s = 128; s > 0; s >>= 1) {
        if (tid < s) { redx[tid] += redx[tid + s]; redy[tid] += redy[tid + s]; }
        __syncthreads();
    }
    if (tid == 0)
        out[n] = redx[0] / (float)CH_P1 + redy[0] / (float)CH_P2;
}

// ---------------------------------------------------------------------------
extern "C" void kernel_launch(void* const* d_in, const int* in_sizes, int n_in,
                              void* d_out, int out_size, void* d_ws, size_t ws_size,
                              hipStream_t stream) {
    const float* x = (const float*)d_in[0];   // [N, P1, 3] fp32
    const float* y = (const float*)d_in[1];   // [N, P2, 3] fp32
    float* out = (float*)d_out;               // [N] fp32

    // Workspace layout (all fp32/u32):
    //   rowmin  [N*P1]      128 KB
    //   colmin  [N*P2]      128 KB
    //   packLo  [N*P2*3]    384 KB
    //   packHi  [N*P2*3]    384 KB
    char* w = (char*)d_ws;
    float*    rowmin = (float*)w;                 w += sizeof(float) * CH_N * CH_P1;
    unsigned* colmin = (unsigned*)w;              w += sizeof(unsigned) * CH_N * CH_P2;
    float*    packLo = (float*)w;                 w += sizeof(float) * CH_N * CH_P2 * 3;
    float*    packHi = (float*)w;

    const int ncol = CH_N * CH_P2;
    chamfer_pack_y<<<(ncol + 255) / 256, 256, 0, stream>>>(y, packLo, packHi, colmin);

    dim3 grid(CH_P1 / ROWS_PER_BLOCK, CH_N);
    chamfer_tiles<<<grid, THREADS, 0, stream>>>(x, packLo, packHi, rowmin, colmin);

    chamfer_finalize<<<CH_N, 256, 0, stream>>>(rowmin, colmin, out);
}